// BIGCF_20684562498310
// MI455X (gfx1250) — compile-verified
//
#include <hip/hip_runtime.h>

#define N_USERS 100000
#define N_ITEMS 50000
#define N_NODES 150000
#define N_EDGES 3200000
#define DIM 64
#define ALPHA 0.5f

typedef float v2f __attribute__((ext_vector_type(2)));
typedef float v8f __attribute__((ext_vector_type(8)));

// ---------------------------------------------------------------------------
// Phase 1: fused intent attention.
//   logits = X @ W + b  (VALU, per-lane)
//   attn   = softmax(logits) over 4     (pair-exchange via __shfl_xor 16)
//   fused  = X + (ALPHA*attn) @ intents (v_wmma_f32_16x16x4_f32, C = X tile)
// One wave32 per 16-row tile. Lane l owns row l%16, K-pair {2*(l/16), +1},
// which is exactly the CDNA5 16x4 f32 A-matrix VGPR layout.
// ---------------------------------------------------------------------------
__global__ __launch_bounds__(32) void bigcf_fuse_kernel(
    const float* __restrict__ user_emb, const float* __restrict__ item_emb,
    const float* __restrict__ user_intents, const float* __restrict__ item_intents,
    const float* __restrict__ Wu, const float* __restrict__ bu,
    const float* __restrict__ Wi, const float* __restrict__ bi,
    float* __restrict__ xbuf, float* __restrict__ accbuf)
{
    __shared__ float tile[16 * DIM];

    const int t  = blockIdx.x;     // tile index 0..9374 (wave-uniform)
    const int l  = threadIdx.x;    // lane 0..31
    const int r0 = t * 16;         // first global node row of this tile
    const bool isUser = (r0 < N_USERS);   // uniform: 100000 % 16 == 0

    const float* __restrict__ emb =
        isUser ? (user_emb + (size_t)r0 * DIM)
               : (item_emb + (size_t)(r0 - N_USERS) * DIM);
    const float* __restrict__ W       = isUser ? Wu : Wi;
    const float* __restrict__ b       = isUser ? bu : bi;
    const float* __restrict__ intents = isUser ? user_intents : item_intents;

    // Cooperative, coalesced load of the 16x64 f32 tile (256 float4s, 8/lane).
    #pragma unroll
    for (int p = 0; p < 8; ++p) {
        int f   = p * 32 + l;        // float4 index within tile
        int row = f >> 4;
        int c4  = (f & 15) << 2;
        *(float4*)(tile + row * DIM + c4) = *(const float4*)(emb + row * DIM + c4);
    }
    __syncthreads();

    // logits: lane l -> row = l%16, K-pair k0 = 2*(l/16)
    const int row = l & 15;
    const int k0  = (l >> 4) << 1;   // 0 for lanes 0-15, 2 for lanes 16-31
    float s0 = b[k0], s1 = b[k0 + 1];
    #pragma unroll 8
    for (int c = 0; c < DIM; ++c) {
        float  xv = tile[row * DIM + c];
        float2 wv = *(const float2*)(W + c * 4 + k0);   // W is [64,4] row-major
        s0 = fmaf(xv, wv.x, s0);
        s1 = fmaf(xv, wv.y, s1);
    }

    // softmax over the 4 logits of this row (partner lane = l ^ 16)
    float m = fmaxf(s0, s1);
    m = fmaxf(m, __shfl_xor(m, 16));
    float e0 = __expf(s0 - m);
    float e1 = __expf(s1 - m);
    float sum = e0 + e1;
    sum += __shfl_xor(sum, 16);
    const float inv = ALPHA / sum;   // fold ALPHA into A

    // A-matrix (16x4 f32): VGPR0 = K0/K2 halves, VGPR1 = K1/K3 halves --
    // exactly (e0*inv, e1*inv) in this lane mapping.
    v2f A;
    A.x = e0 * inv;
    A.y = e1 * inv;

    const int n     = l & 15;           // N index for B/C/D
    const int mbase = (l >> 4) << 3;    // C/D rows: M = p + (0|8)

    #pragma unroll
    for (int j = 0; j < 4; ++j) {       // four 16x16 output tiles across d=64
        // B-matrix (4x16 f32): VGPR0 = K0/K2, VGPR1 = K1/K3, N across lanes
        v2f B;
        B.x = intents[(k0 + 0) * DIM + j * 16 + n];
        B.y = intents[(k0 + 1) * DIM + j * 16 + n];

        // C = embedding tile in the 16x16 f32 C/D layout
        v8f C;
        #pragma unroll
        for (int p = 0; p < 8; ++p)
            C[p] = tile[(p + mbase) * DIM + j * 16 + n];

        // D = (ALPHA*attn) @ intents + X
        v8f D = __builtin_amdgcn_wmma_f32_16x16x4_f32(
            false, A, false, B, (short)0, C, false, false);

        #pragma unroll
        for (int p = 0; p < 8; ++p) {
            int idx = (r0 + p + mbase) * DIM + j * 16 + n;
            xbuf[idx]   = D[p];   // x_0 for propagation
            accbuf[idx] = D[p];   // acc init
        }
    }
}

// ---------------------------------------------------------------------------
// Phase 2: edge-parallel SpMM  y[r,:] += v * x[c,:].
// One wave per edge, 2 columns per lane (coalesced 128B gather + atomics).
// x/y/acc (~115 MB total) are L2-resident on MI455X (192 MB L2), so the
// random gathers and the non-returning global_atomic_add_f32 resolve in L2.
// ---------------------------------------------------------------------------
__global__ __launch_bounds__(256) void bigcf_spmm_kernel(
    const float* __restrict__ x, const int* __restrict__ rows,
    const int* __restrict__ cols, const float* __restrict__ vals,
    float* __restrict__ y)
{
    unsigned t = blockIdx.x * 256u + threadIdx.x;
    int e = (int)(t >> 5);
    int l = (int)(t & 31u);
    if (e >= N_EDGES) return;
    int   r = rows[e];
    int   c = cols[e];
    float v = vals[e];
    float x0 = x[(size_t)c * DIM + l];
    float x1 = x[(size_t)c * DIM + l + 32];
    atomicAdd(y + (size_t)r * DIM + l,      v * x0);
    atomicAdd(y + (size_t)r * DIM + l + 32, v * x1);
}

// acc = (acc + y) * scale   (scale = 1, 1, 0.25 across the three layers)
__global__ __launch_bounds__(256) void bigcf_accum_kernel(
    float* __restrict__ acc, const float* __restrict__ y, float scale, int n)
{
    int i = blockIdx.x * 256 + threadIdx.x;
    if (i < n) acc[i] = (acc[i] + y[i]) * scale;
}

extern "C" void kernel_launch(void* const* d_in, const int* in_sizes, int n_in,
                              void* d_out, int out_size, void* d_ws, size_t ws_size,
                              hipStream_t stream)
{
    const float* user_emb     = (const float*)d_in[0];
    const float* item_emb     = (const float*)d_in[1];
    const float* user_intents = (const float*)d_in[2];
    const float* item_intents = (const float*)d_in[3];
    const float* Wu           = (const float*)d_in[4];
    const float* bu           = (const float*)d_in[5];
    const float* Wi           = (const float*)d_in[6];
    const float* bi           = (const float*)d_in[7];
    const float* adj_vals     = (const float*)d_in[8];
    const int*   adj_rows     = (const int*)d_in[9];
    const int*   adj_cols     = (const int*)d_in[10];
    float* out = (float*)d_out;

    const size_t nelems = (size_t)N_NODES * DIM;       // 9.6M floats
    float* xb = (float*)d_ws;                          // ping
    float* yb = xb + nelems;                           // pong (ws >= 76.8 MB)

    // Phase 1: fused attention -> xb (x_0) and out (acc)
    bigcf_fuse_kernel<<<N_NODES / 16, 32, 0, stream>>>(
        user_emb, item_emb, user_intents, item_intents,
        Wu, bu, Wi, bi, xb, out);

    const int spmmBlocks = (N_EDGES * 32 + 255) / 256; // 400000
    const int accBlocks  = (int)((nelems + 255) / 256);

    float* cur = xb;
    float* nxt = yb;
    for (int layer = 0; layer < 3; ++layer) {
        hipMemsetAsync(nxt, 0, nelems * sizeof(float), stream);
        bigcf_spmm_kernel<<<spmmBlocks, 256, 0, stream>>>(
            cur, adj_rows, adj_cols, adj_vals, nxt);
        bigcf_accum_kernel<<<accBlocks, 256, 0, stream>>>(
            out, nxt, (layer == 2) ? 0.25f : 1.0f, (int)nelems);
        float* tmp = cur; cur = nxt; nxt = tmp;
    }
}